// DFFormerMoELoss_54348516163751
// MI455X (gfx1250) — compile-verified
//
#include <hip/hip_runtime.h>
#include <hip/hip_bf16.h>
#include <math.h>

// ---------------------------------------------------------------------------
// DFFormer MoE ranking loss for MI455X (gfx1250, wave32).
//   out = 1.0*listMLE + 0.5*pairwise_hinge + 0.01*load_bal
// B=16 rows, N=2048 cols, f32. Latency-bound problem; all data LDS/L2
// resident. WMMA f32 16x16x4 with all-ones B is used as the full-precision
// 32-lane reduction engine (tensor-core reduction).
// ---------------------------------------------------------------------------

#define BB 16
#define NN 2048
#define NTHREADS 256
#define NPT (NN / NTHREADS)      // 8 elements per thread for scans
#define JCHUNK 128               // pairwise j-tile per block
#define NCHUNK (NN / JCHUNK)     // 16 chunks per row
#define MARGIN 0.1f

typedef __attribute__((ext_vector_type(2))) float v2f;
typedef __attribute__((ext_vector_type(8))) float v8f;

// Full 32-lane f32 sum via V_WMMA_F32_16X16X4_F32 with all-ones B.
// A holds one partial per lane (second slot zero); D = A x ones, so every
// column of D sums all A slots. Per-lane sum of the 8 C/D VGPRs gives the
// half-column sums (C layout: VGPR r = rows r / r+8 split by lane half);
// shfl_xor(16) completes the total in every lane. EXEC must be all ones.
__device__ __forceinline__ float wave_reduce_wmma(float x) {
  v2f a; a.x = x;    a.y = 0.0f;
  v2f b; b.x = 1.0f; b.y = 1.0f;
  v8f c = {};
  v8f d = __builtin_amdgcn_wmma_f32_16x16x4_f32(
      /*neg_a=*/false, a, /*neg_b=*/false, b,
      /*c_mod=*/(short)0, c, /*reuse_a=*/false, /*reuse_b=*/false);
  float s = d[0] + d[1] + d[2] + d[3] + d[4] + d[5] + d[6] + d[7];
  s += __shfl_xor(s, 16, 32);
  return s;
}

// ---------------------------------------------------------------------------
// Kernel A: per-row listMLE. One block per row.
//  - bitonic sort (key = masked y_true desc, tie = index asc  ==  stable
//    argsort(-mt)), payload = masked y_pred
//  - suffix cumulative logsumexp via per-thread chunks + Hillis-Steele scan
//    over (max, sumexp) aggregates
//  - ll reduced with the WMMA lane-reduction
// ---------------------------------------------------------------------------
__global__ __launch_bounds__(NTHREADS)
void listmle_kernel(const float* __restrict__ yp, const float* __restrict__ yt,
                    const float* __restrict__ mk, float* __restrict__ out_listmle) {
  __shared__ float k_s[NN];   // keys (mt)
  __shared__ float v_s[NN];   // payload (mp)
  __shared__ int   i_s[NN];   // tie-break index
  __shared__ float am[NTHREADS], as_[NTHREADS], part[NTHREADS];

  const int row = blockIdx.x;
  const int tid = threadIdx.x;
  const float* rp = yp + row * NN;
  const float* rt = yt + row * NN;
  const float* rm = mk + row * NN;

  for (int t = tid; t < NN; t += NTHREADS) {
    float m = rm[t];
    k_s[t] = rt[t] * m;
    v_s[t] = rp[t] * m;
    i_s[t] = t;
  }
  __syncthreads();

  // Bitonic sort, final order: key descending, index ascending on ties.
  for (int k = 2; k <= NN; k <<= 1) {
    for (int j = k >> 1; j > 0; j >>= 1) {
      for (int t = tid; t < NN; t += NTHREADS) {
        int ixj = t ^ j;
        if (ixj > t) {
          float ka = k_s[t], kb = k_s[ixj];
          int   ia = i_s[t], ib = i_s[ixj];
          // b_first: element at ixj should precede element at t
          bool b_first = (kb > ka) || (kb == ka && ib < ia);
          bool asc = ((t & k) == 0);
          if (asc ? b_first : !b_first) {
            float va = v_s[t], vb = v_s[ixj];
            k_s[t] = kb; k_s[ixj] = ka;
            v_s[t] = vb; v_s[ixj] = va;
            i_s[t] = ib; i_s[ixj] = ia;
          }
        }
      }
      __syncthreads();
    }
  }

  // Chunk aggregate (max, sum of exp) for suffix logsumexp.
  const int base = tid * NPT;
  float cm = -INFINITY;
  for (int k = 0; k < NPT; ++k) cm = fmaxf(cm, v_s[base + k]);
  float cs = 0.f;
  for (int k = 0; k < NPT; ++k) cs += expf(v_s[base + k] - cm);
  am[tid] = cm; as_[tid] = cs;
  __syncthreads();

  // Inclusive suffix scan over chunk aggregates (Hillis-Steele).
  float sm = cm, ss = cs;
  for (int d = 1; d < NTHREADS; d <<= 1) {
    float om = -INFINITY, os = 0.f;
    if (tid + d < NTHREADS) { om = am[tid + d]; os = as_[tid + d]; }
    __syncthreads();
    float M = fmaxf(sm, om);
    ss = ss * expf(sm - M) + os * expf(om - M);
    sm = M;
    am[tid] = sm; as_[tid] = ss;
    __syncthreads();
  }

  // Exclusive suffix aggregate for this chunk.
  float em = -INFINITY, es = 0.f;
  if (tid + 1 < NTHREADS) { em = am[tid + 1]; es = as_[tid + 1]; }

  // Walk the chunk back-to-front; last global term is exactly 0 so summing
  // all j matches the reference's j < n-1 sum.
  float rmax = em, rsum = es, ll = 0.f;
  for (int k = NPT - 1; k >= 0; --k) {
    float v = v_s[base + k];
    float M = fmaxf(rmax, v);
    rsum = rsum * expf(rmax - M) + expf(v - M);
    rmax = M;
    ll += v - (rmax + logf(rsum));
  }
  part[tid] = ll;
  __syncthreads();

  if (tid < 32) {                       // wave 0, EXEC all ones
    float acc = 0.f;
    for (int w = 0; w < NTHREADS / 32; ++w) acc += part[tid + 32 * w];
    float tot = wave_reduce_wmma(acc);
    if (tid == 0) out_listmle[row] = -tot / (float)NN;
  }
}

// ---------------------------------------------------------------------------
// Kernel B: pairwise hinge partials. Grid = (chunk, row). Each block covers
// all i against a 128-wide j tile held in LDS; WMMA lane-reduction at the
// end, partials written to unique ws slots (deterministic, no atomics).
// ---------------------------------------------------------------------------
__global__ __launch_bounds__(NTHREADS)
void pairwise_kernel(const float* __restrict__ yp, const float* __restrict__ yt,
                     const float* __restrict__ mk,
                     float* __restrict__ ws_sum, float* __restrict__ ws_cnt) {
  __shared__ float jt[JCHUNK], jp[JCHUNK];
  __shared__ float psum[NTHREADS], pcnt[NTHREADS];

  const int chunk = blockIdx.x;
  const int row   = blockIdx.y;
  const int tid   = threadIdx.x;
  const float* rp = yp + row * NN;
  const float* rt = yt + row * NN;
  const float* rm = mk + row * NN;

  const int j0 = chunk * JCHUNK;
  if (tid < JCHUNK) {
    float m = rm[j0 + tid];
    jp[tid] = rp[j0 + tid] * m;
    jt[tid] = rt[j0 + tid] * m;
  }
  __syncthreads();

  float hs = 0.f, hc = 0.f;
  for (int i = tid; i < NN; i += NTHREADS) {
    float m  = rm[i];
    float pi = rp[i] * m;
    float ti = rt[i] * m;
#pragma unroll 4
    for (int j = 0; j < JCHUNK; ++j) {
      float td = ti - jt[j];
      float h  = fmaxf(MARGIN - (pi - jp[j]), 0.f);
      bool  g  = td > 0.f;
      hs += g ? h : 0.f;
      hc += g ? 1.f : 0.f;
    }
  }
  psum[tid] = hs; pcnt[tid] = hc;
  __syncthreads();

  if (tid < 32) {                       // wave 0, EXEC all ones
    float a1 = 0.f, a2 = 0.f;
    for (int w = 0; w < NTHREADS / 32; ++w) {
      a1 += psum[tid + 32 * w];
      a2 += pcnt[tid + 32 * w];
    }
    float s = wave_reduce_wmma(a1);
    float c = wave_reduce_wmma(a2);
    if (tid == 0) {
      ws_sum[row * NCHUNK + chunk] = s;
      ws_cnt[row * NCHUNK + chunk] = c;
    }
  }
}

// ---------------------------------------------------------------------------
// Kernel C: single-wave deterministic combine.
// ---------------------------------------------------------------------------
__global__ __launch_bounds__(32)
void combine_kernel(const float* __restrict__ ws_listmle,
                    const float* __restrict__ ws_sum,
                    const float* __restrict__ ws_cnt,
                    const float* __restrict__ lbl,
                    float* __restrict__ out) {
  const int tid = threadIdx.x;

  float lm = (tid < BB) ? ws_listmle[tid] : 0.f;
  float ranking = wave_reduce_wmma(lm) / (float)BB;

  float pb = 0.f;
  if (tid < BB) {
    float s = 0.f, c = 0.f;
    for (int ch = 0; ch < NCHUNK; ++ch) {   // fixed order: deterministic
      s += ws_sum[tid * NCHUNK + ch];
      c += ws_cnt[tid * NCHUNK + ch];
    }
    pb = s / (c + 1e-12f);
  }
  float pmean = wave_reduce_wmma(pb) / (float)BB;

  if (tid == 0)
    out[0] = 1.0f * ranking + 0.5f * pmean + 0.01f * lbl[0];
}

// ---------------------------------------------------------------------------
extern "C" void kernel_launch(void* const* d_in, const int* in_sizes, int n_in,
                              void* d_out, int out_size, void* d_ws, size_t ws_size,
                              hipStream_t stream) {
  const float* yp  = (const float*)d_in[0];   // y_pred  [16,2048] f32
  const float* yt  = (const float*)d_in[1];   // y_true  [16,2048] f32
  const float* mk  = (const float*)d_in[2];   // masks   [16,2048] f32
  const float* lbl = (const float*)d_in[3];   // load_bal_loss scalar f32
  float* out = (float*)d_out;

  float* ws          = (float*)d_ws;
  float* ws_listmle  = ws;                    // 16 floats
  float* ws_sum      = ws + BB;               // 16*16 floats
  float* ws_cnt      = ws + BB + BB * NCHUNK; // 16*16 floats

  listmle_kernel<<<BB, NTHREADS, 0, stream>>>(yp, yt, mk, ws_listmle);
  pairwise_kernel<<<dim3(NCHUNK, BB), NTHREADS, 0, stream>>>(yp, yt, mk, ws_sum, ws_cnt);
  combine_kernel<<<1, 32, 0, stream>>>(ws_listmle, ws_sum, ws_cnt, lbl, out);
}